// Router_7155415515698
// MI455X (gfx1250) — compile-verified
//
#include <hip/hip_runtime.h>

// Problem constants (match reference)
#define B_ 4
#define S_ 4096
#define D_ 4096
#define E_ 64
#define CAP_ 128

typedef __attribute__((ext_vector_type(2))) float v2f;
typedef __attribute__((ext_vector_type(8))) float v8f;

// LDS row stride in floats: 132 => rows 16B aligned, and fragment reads hit
// bank (4*m + c) % 64 -> conflict-free across the 16-lane groups.
#define LDS_STRIDE 132

// ---------------------------------------------------------------------------
// Kernel 1: logits[M=16384, E=64] = x[M, D] @ W[E, D]^T  via V_WMMA_F32_16X16X4_F32
// Block: 128 threads (4 waves). Block owns 16 rows x 64 experts.
// Wave w computes the 16x16 tile for experts [16w, 16w+16).
// K tiled by 128 through LDS with coalesced float4 staging.
// ---------------------------------------------------------------------------
__global__ __launch_bounds__(128) void router_gemm_wmma(
    const float* __restrict__ x, const float* __restrict__ W,
    float* __restrict__ logits) {
  __shared__ float xs[16 * LDS_STRIDE];   // 16 rows  x 128 K (padded)
  __shared__ float ws[64 * LDS_STRIDE];   // 64 exprt x 128 K (padded)

  const int t    = threadIdx.x;
  const int lane = t & 31;
  const int wave = t >> 5;
  const int row0 = blockIdx.x * 16;
  const int eb   = wave * 16;

  // Fragment addressing per ISA 16x4 f32 layout:
  // lanes 0-15 hold K = k,k+1 ; lanes 16-31 hold K = k+2,k+3.
  const int mn   = lane & 15;          // M index (A) / N index (B)
  const int ksel = (lane >> 4) * 2;    // 0 or 2

  v8f acc = {};  // 16x16 f32 accumulator (8 VGPRs)

  for (int kb = 0; kb < D_; kb += 128) {
    // Stage x tile: 16x128 floats = 512 float4, 128 threads x 4 iters.
    #pragma unroll
    for (int i = 0; i < 4; ++i) {
      const int off = (t + i * 128) * 4;          // flat float offset
      const int r = off >> 7, c = off & 127;
      const float4 v = *(const float4*)(x + (size_t)(row0 + r) * D_ + kb + c);
      *(float4*)(&xs[r * LDS_STRIDE + c]) = v;
    }
    // Stage W tile: 64x128 floats = 2048 float4, 128 threads x 16 iters.
    #pragma unroll
    for (int i = 0; i < 16; ++i) {
      const int off = (t + i * 128) * 4;
      const int r = off >> 7, c = off & 127;
      const float4 v = *(const float4*)(W + (size_t)r * D_ + kb + c);
      *(float4*)(&ws[r * LDS_STRIDE + c]) = v;
    }
    __syncthreads();

    // Hint next x tile toward the caches while we compute on this one.
    if (kb + 128 < D_)
      __builtin_prefetch(x + (size_t)(row0 + (t & 15)) * D_ + kb + 128, 0, 1);

    #pragma unroll
    for (int kk = 0; kk < 128; kk += 4) {
      const v2f a = *(const v2f*)(&xs[mn * LDS_STRIDE + kk + ksel]);
      const v2f b = *(const v2f*)(&ws[(eb + mn) * LDS_STRIDE + kk + ksel]);
      acc = __builtin_amdgcn_wmma_f32_16x16x4_f32(
          /*neg_a=*/false, a, /*neg_b=*/false, b,
          /*c_mod=*/(short)0, acc, /*reuse_a=*/false, /*reuse_b=*/false);
    }
    __syncthreads();
  }

  // C/D layout: VGPR r -> (M = r + (lane>=16 ? 8 : 0), N = lane&15)
  const int n  = lane & 15;
  const int mo = (lane >> 4) * 8;
  #pragma unroll
  for (int r = 0; r < 8; ++r)
    logits[(size_t)(row0 + r + mo) * E_ + eb + n] = acc[r];
}

// ---------------------------------------------------------------------------
// Kernel 2: per-row softmax stats over E=64 logits.
//   argmax (first-max tie break, like jnp.argmax) -> idx (workspace)
//   router_prob = max(softmax) = 1 / sum(exp(l - lmax))
// ---------------------------------------------------------------------------
__global__ __launch_bounds__(256) void router_stats(
    const float* __restrict__ logits, float* __restrict__ rprobs,
    int* __restrict__ idx) {
  const int row = blockIdx.x * blockDim.x + threadIdx.x;
  if (row >= B_ * S_) return;
  const float* lg = logits + (size_t)row * E_;

  float mx = lg[0];
  int am = 0;
  #pragma unroll 4
  for (int j = 1; j < E_; ++j) {
    const float v = lg[j];
    if (v > mx) { mx = v; am = j; }
  }
  float s = 0.f;
  #pragma unroll 4
  for (int j = 0; j < E_; ++j) s += expf(lg[j] - mx);

  rprobs[row] = 1.0f / s;
  idx[row] = am;
}

// ---------------------------------------------------------------------------
// Kernel 3: zero-fill expert_indices (d_out is poisoned by the harness).
// ---------------------------------------------------------------------------
__global__ void zero_fill(float4* __restrict__ p, int n4) {
  const int stride = gridDim.x * blockDim.x;
  const float4 z = make_float4(0.f, 0.f, 0.f, 0.f);
  for (int i = blockIdx.x * blockDim.x + threadIdx.x; i < n4; i += stride)
    p[i] = z;
}

// ---------------------------------------------------------------------------
// Kernel 4: capacity-limited one-hot. One wave32 per (batch, expert).
// Sequential-over-S semantics of jnp.cumsum reproduced with ballot + popcount
// prefix ranks; only in-capacity matches write 1.0 (buffer pre-zeroed).
// ---------------------------------------------------------------------------
__global__ __launch_bounds__(32) void router_capacity(
    const int* __restrict__ idx, float* __restrict__ outEI) {
  const int be = blockIdx.x;           // 0 .. B*E-1
  const int b  = be >> 6;              // E_ == 64
  const int e  = be & 63;
  const int lane = threadIdx.x;        // wave32
  const unsigned lt = (1u << lane) - 1u;

  const int* bi = idx + b * S_;
  int carry = 0;
  for (int c = 0; c < S_ / 32; ++c) {
    const int s = c * 32 + lane;
    const bool match = (bi[s] == e);
    const unsigned mask = (unsigned)(__ballot(match) & 0xFFFFFFFFull);
    if (match) {
      const int rank = carry + __popc(mask & lt) + 1;  // 1-based priority
      if (rank <= CAP_)
        outEI[(size_t)(b * S_ + s) * E_ + e] = 1.0f;
    }
    carry += __popc(mask);
  }
}

// ---------------------------------------------------------------------------
extern "C" void kernel_launch(void* const* d_in, const int* in_sizes, int n_in,
                              void* d_out, int out_size, void* d_ws,
                              size_t ws_size, hipStream_t stream) {
  const float* x = (const float*)d_in[0];   // [B,S,D]
  const float* W = (const float*)d_in[1];   // [E,D]

  float* out   = (float*)d_out;
  float* outEI = out;                                   // [B,S,E]
  float* outRP = out + (size_t)B_ * S_ * E_;            // [B,S,1]
  float* outLG = outRP + (size_t)B_ * S_;               // [B,S,E]
  int*   idx   = (int*)d_ws;                            // [B,S]

  // 1) logits GEMM (memory-bound: 256 MB of x @ 23.3 TB/s ~ 11 us floor)
  router_gemm_wmma<<<(B_ * S_) / 16, 128, 0, stream>>>(x, W, outLG);
  // 2) argmax + max-softmax-prob per token
  router_stats<<<(B_ * S_ + 255) / 256, 256, 0, stream>>>(outLG, outRP, idx);
  // 3) zero expert_indices
  zero_fill<<<1024, 256, 0, stream>>>((float4*)outEI, (B_ * S_ * E_) / 4);
  // 4) capacity-masked one-hot
  router_capacity<<<B_ * E_, 32, 0, stream>>>(idx, outEI);
}